// ThresholdLayer_69243462746427
// MI455X (gfx1250) — compile-verified
//
#include <hip/hip_runtime.h>
#include <cstdint>

#define IN_DIM  4096
#define OUT_DIM 8192
#define NACT    32
#define BATCH   1024

// ---- CDNA5 async global->LDS DMA (per-lane LDS scatter => free transpose) ----
__device__ __forceinline__ void async_gl_to_lds_b32(unsigned lds_off, const void* gaddr) {
    unsigned long long ga = (unsigned long long)gaddr;
    asm volatile("global_load_async_to_lds_b32 %0, %1, off"
                 :: "v"(lds_off), "v"(ga) : "memory");
}
__device__ __forceinline__ void wait_async0() {
    asm volatile("s_wait_asynccnt 0" ::: "memory");
}
__device__ __forceinline__ unsigned lds_off_of(const void* p) {
    // gfx1250 generic LDS address: low 32 bits are the wave-relative LDS offset
    return (unsigned)(unsigned long long)(uintptr_t)p;
}

// ---- K1: x[1024][4096] -> xT[4096][1024], async-DMA transposed into LDS ----
__global__ __launch_bounds__(256) void transpose_x(const float* __restrict__ x,
                                                   float* __restrict__ xT) {
    __shared__ float tile[32][33];
    const int tx = threadIdx.x;           // 0..31
    const int ty = threadIdx.y;           // 0..7
    const int c0 = blockIdx.x * 32;
    const int b0 = blockIdx.y * 32;
#pragma unroll
    for (int j = 0; j < 32; j += 8) {
        const float* g = x + (size_t)(b0 + ty + j) * IN_DIM + c0 + tx;
        async_gl_to_lds_b32(lds_off_of(&tile[tx][ty + j]), g);  // transposed LDS write
    }
    wait_async0();
    __syncthreads();
#pragma unroll
    for (int j = 0; j < 32; j += 8)
        xT[(size_t)(c0 + ty + j) * BATCH + b0 + tx] = tile[ty + j][tx];
}

// ---- K2 (fused): 16 waves x 16 rows x 128 batches per block.
//   wave w -> row r0+w; lane l -> batches b0+4l..4l+3 (float4)
//   taps (idx,w) held in SGPRs via readfirstlane-uniform row id
//   result staged in LDS [16][132] then stored transposed with bias fused.
#define ROWS_PB   16
#define BATCH_PB  128
#define LDS_STRIDE 132   // 132*4 = 528 B row stride: 16B aligned, conflict-free b128 writes

__global__ __launch_bounds__(512) void sparse_dot_store(const long long* __restrict__ idx,
                                                        const float* __restrict__ w,
                                                        const float* __restrict__ bias,
                                                        const float* __restrict__ xT,
                                                        float* __restrict__ y) {
    __shared__ float tile[ROWS_PB * LDS_STRIDE];

    const int lane = threadIdx.x & 31;
    const int wv_id = __builtin_amdgcn_readfirstlane((int)(threadIdx.x >> 5)); // 0..15
    const int r0 = blockIdx.x * ROWS_PB;
    const int b0 = blockIdx.y * BATCH_PB;
    const int r  = r0 + wv_id;                       // wave-uniform row

    const long long* ip = idx + (long long)r * NACT;
    const float*     wp = w   + (size_t)r * NACT;

    int   col[NACT];
    float wt[NACT];
#pragma unroll
    for (int t = 0; t < NACT; ++t) { col[t] = (int)ip[t]; wt[t] = wp[t]; }

    float ax = 0.f, ay = 0.f, az = 0.f, aw = 0.f;
#pragma unroll
    for (int t = 0; t < NACT; ++t) {
        const float4* p = (const float4*)(xT + (size_t)col[t] * BATCH + b0);
        float4 v = p[lane];                          // 512B contiguous per wave (L2-resident)
        ax = fmaf(wt[t], v.x, ax);
        ay = fmaf(wt[t], v.y, ay);
        az = fmaf(wt[t], v.z, az);
        aw = fmaf(wt[t], v.w, aw);
    }

    // stage: tile[row][b_local], b128 per lane, conflict-free
    float4* trow = (float4*)(tile + wv_id * LDS_STRIDE);
    trow[lane] = make_float4(ax, ay, az, aw);
    __syncthreads();

    // store phase: thread t -> batch bi = t>>2, rows rj*4..rj*4+3 (rj = t&3)
    const int bi = threadIdx.x >> 2;                 // 0..127
    const int rj = threadIdx.x & 3;                  // 0..3  -> 16 rows as float4
    const float4 b4 = *(const float4*)(bias + r0 + rj * 4);
    float4 o;
    o.x = tile[(rj * 4 + 0) * LDS_STRIDE + bi] - b4.x;
    o.y = tile[(rj * 4 + 1) * LDS_STRIDE + bi] - b4.y;
    o.z = tile[(rj * 4 + 2) * LDS_STRIDE + bi] - b4.z;
    o.w = tile[(rj * 4 + 3) * LDS_STRIDE + bi] - b4.w;
    *(float4*)(y + (size_t)(b0 + bi) * OUT_DIM + r0 + rj * 4) = o;  // 64B segments, coalesced
}

extern "C" void kernel_launch(void* const* d_in, const int* in_sizes, int n_in,
                              void* d_out, int out_size, void* d_ws, size_t ws_size,
                              hipStream_t stream) {
    (void)in_sizes; (void)n_in; (void)out_size; (void)ws_size;
    const float*     x    = (const float*)d_in[0];
    const float*     wk   = (const float*)d_in[1];
    const float*     bias = (const float*)d_in[2];
    const long long* idx  = (const long long*)d_in[3];   // reference dtype int64
    float* y  = (float*)d_out;

    float* xT = (float*)d_ws;                            // 16 MB workspace only

    transpose_x<<<dim3(IN_DIM / 32, BATCH / 32), dim3(32, 8), 0, stream>>>(x, xT);
    sparse_dot_store<<<dim3(OUT_DIM / ROWS_PB, BATCH / BATCH_PB), 512, 0, stream>>>(
        idx, wk, bias, xT, y);
}